// TwoWayMultiLabelLoss_60653528154207
// MI455X (gfx1250) — compile-verified
//
#include <hip/hip_runtime.h>
#include <hip/hip_bf16.h>

#define B_ROWS 4096
#define C_COLS 5000
#define NBLK   256
#define NCHUNK 16                      // row chunks (256 rows each) for column partials
#define CCHUNKS 20                     // col chunks (256 cols each, last has 136 live)
#define TROWS  16                      // rows per async-staged tile
#define NTILE  (256 / TROWS)           // 16 tiles per block
#define FLT_BIG_NEG (-3.402823466e38f)

static __device__ __forceinline__ void lse_merge(float& m, float& s, float m2, float s2) {
    // branchless pair merge; identity state (m=-FLT_MAX, s=0) is safe (exp(0)=1, s=0)
    float M = fmaxf(m, m2);
    s = s * __expf(m - M) + s2 * __expf(m2 - M);
    m = M;
}

static __device__ __forceinline__ void lse_push(float& m, float& s, float x) {
    // branch form: exactly one v_exp on either path (vs two in the fused form)
    if (x > m) { s = s * __expf(m - x) + 1.0f; m = x; }
    else       { s += __expf(x - m); }
}

static __device__ __forceinline__ float two_way_loss(float m_n, float s_n,
                                                     float m_p, float s_p,
                                                     int cnt_p, int cnt_n) {
    if (cnt_p <= 0 || cnt_n <= 0) return 0.0f;
    float lse_neg = m_n + logf(s_n);            // T_N = 1
    float lse_pos = 4.0f * (m_p + logf(s_p));   // T_P = 4, accumulated on -x/4
    float z = lse_neg + lse_pos;
    return (z > 0.0f) ? (z + log1pf(expf(-z))) : log1pf(expf(z));   // stable softplus
}

// ---------------------------------------------------------------------------
// Fused single-pass kernel: block = 256 rows x 256 cols tile of the matrix.
// Thread owns one column (register LSE state); per-row state is built by a
// wave32 shfl_xor butterfly + LDS cross-wave merge per 16-row async tile.
// Data is staged global->LDS with the CDNA5 async DMA path, double buffered.
// ---------------------------------------------------------------------------
__global__ __launch_bounds__(NBLK) void fused_partial_kernel(
        const float* __restrict__ logits, const int* __restrict__ targets,
        float* __restrict__ rpm_n, float* __restrict__ rps_n,
        float* __restrict__ rpm_p, float* __restrict__ rps_p, int* __restrict__ rpc,
        float* __restrict__ cpm_n, float* __restrict__ cps_n,
        float* __restrict__ cpm_p, float* __restrict__ cps_p, int* __restrict__ cpc) {
    __shared__ __align__(16) float tileL[2][TROWS][NBLK];   // 32 KB, b128 DMA dest
    __shared__ __align__(16) int   tileT[2][TROWS][NBLK];   // 32 KB, b128 DMA dest
    __shared__ float wrm_n[8][TROWS], wrs_n[8][TROWS];
    __shared__ float wrm_p[8][TROWS], wrs_p[8][TROWS];
    __shared__ int   wrc[8][TROWS];

    const int tid    = threadIdx.x;
    const int cchunk = blockIdx.x;
    const int rchunk = blockIdx.y;
    const int c0     = cchunk * NBLK;
    const int r0     = rchunk * NBLK;
    const int wave   = tid >> 5, lane = tid & 31;
    const bool validc = (c0 + tid) < C_COLS;

    // 4 b128 async DMAs per array per thread-slot covers a 16x256 f32 tile.
    auto issueTile = [&](int buf, int t) {
        const int rbase = r0 + t * TROWS;
#pragma unroll
        for (int i = 0; i < 4; ++i) {
            int lin = i * 1024 + tid * 4;           // linear index inside 16x256 tile
            int rr = lin >> 8;                      // tile row 0..15
            int cc = lin & 255;                     // tile col, multiple of 4
            int gc = c0 + cc;
            if (gc > C_COLS - 4) gc = C_COLS - 4;   // clamp: in-bounds + 16B aligned
            size_t gidx = (size_t)(rbase + rr) * C_COLS + gc;
            const float* gl = logits  + gidx;
            const int*   gt = targets + gidx;
            unsigned dl = (unsigned)(size_t)&tileL[buf][rr][cc];
            unsigned dt = (unsigned)(size_t)&tileT[buf][rr][cc];
            asm volatile("global_load_async_to_lds_b128 %0, %1, off" :: "v"(dl), "v"(gl) : "memory");
            asm volatile("global_load_async_to_lds_b128 %0, %1, off" :: "v"(dt), "v"(gt) : "memory");
        }
    };

    // column (per-class) accumulators, register resident
    float m_n = FLT_BIG_NEG, s_n = 0.0f, m_p = FLT_BIG_NEG, s_p = 0.0f;
    int cp = 0, cn = 0;

    issueTile(0, 0);
    for (int t = 0; t < NTILE; ++t) {
        const int buf = t & 1;
        if (t + 1 < NTILE) {
            issueTile(buf ^ 1, t + 1);
            asm volatile("s_wait_asynccnt 0x8" ::: "memory");  // 8 newest pending -> prev tile done
        } else {
            asm volatile("s_wait_asynccnt 0x0" ::: "memory");
        }
        __syncthreads();   // all waves' DMA data visible

#pragma unroll
        for (int j = 0; j < TROWS; ++j) {
            float x  = tileL[buf][j][tid];
            int   tv = tileT[buf][j][tid];
            bool isp = validc && (tv == 1);
            bool isn = validc && (tv != 1);

            // column state
            if (isp)      { lse_push(m_p, s_p, -0.25f * x); }
            else if (isn) { lse_push(m_n, s_n, x); }
            cp += isp ? 1 : 0;
            cn += isn ? 1 : 0;

            // row state: single-element leaf, butterfly all-reduce across wave32
            float lmn = isn ? x : FLT_BIG_NEG;          float lsn = isn ? 1.0f : 0.0f;
            float lmp = isp ? -0.25f * x : FLT_BIG_NEG; float lsp = isp ? 1.0f : 0.0f;
            int   lc  = (isp ? 0x10000 : 0) + (isn ? 1 : 0);
#pragma unroll
            for (int off = 16; off > 0; off >>= 1) {
                float mm = __shfl_xor(lmn, off), ss = __shfl_xor(lsn, off);
                lse_merge(lmn, lsn, mm, ss);
                mm = __shfl_xor(lmp, off); ss = __shfl_xor(lsp, off);
                lse_merge(lmp, lsp, mm, ss);
                lc += __shfl_xor(lc, off);
            }
            if (lane == 0) {
                wrm_n[wave][j] = lmn; wrs_n[wave][j] = lsn;
                wrm_p[wave][j] = lmp; wrs_p[wave][j] = lsp;
                wrc[wave][j] = lc;
            }
        }
        __syncthreads();   // wave row-partials complete; tile buffer fully consumed

        if (tid < TROWS) { // 16 threads finish the 8-wave merge for this tile's rows
            float Mn = wrm_n[0][tid], Sn = wrs_n[0][tid];
            float Mp = wrm_p[0][tid], Sp = wrs_p[0][tid];
            int   Cc = wrc[0][tid];
#pragma unroll
            for (int w = 1; w < 8; ++w) {
                lse_merge(Mn, Sn, wrm_n[w][tid], wrs_n[w][tid]);
                lse_merge(Mp, Sp, wrm_p[w][tid], wrs_p[w][tid]);
                Cc += wrc[w][tid];
            }
            const size_t o = (size_t)cchunk * B_ROWS + (r0 + t * TROWS + tid);
            rpm_n[o] = Mn; rps_n[o] = Sn; rpm_p[o] = Mp; rps_p[o] = Sp; rpc[o] = Cc;
        }
        // no extra barrier needed: next tile's post-wait __syncthreads orders
        // waveRow reuse, and the consumed buffer is only rewritten by DMAs
        // issued after this point.
    }

    if (validc) {
        const size_t o = (size_t)rchunk * C_COLS + (c0 + tid);
        cpm_n[o] = m_n; cps_n[o] = s_n; cpm_p[o] = m_p; cps_p[o] = s_p;
        cpc[o] = (cp << 16) | cn;
    }
}

// ---------------------------------------------------------------------------
// Merge 20 col-chunk partials per row (fixed order) -> per-row loss.
// ---------------------------------------------------------------------------
__global__ __launch_bounds__(NBLK) void row_combine_kernel(
        const float* __restrict__ rpm_n, const float* __restrict__ rps_n,
        const float* __restrict__ rpm_p, const float* __restrict__ rps_p,
        const int* __restrict__ rpc, float* __restrict__ rowLoss) {
    const int row = blockIdx.x * NBLK + threadIdx.x;   // grid covers exactly 4096
    float Mn = FLT_BIG_NEG, Sn = 0.0f, Mp = FLT_BIG_NEG, Sp = 0.0f;
    int pc = 0;
#pragma unroll
    for (int ch = 0; ch < CCHUNKS; ++ch) {
        const size_t o = (size_t)ch * B_ROWS + row;
        lse_merge(Mn, Sn, __builtin_nontemporal_load(&rpm_n[o]), __builtin_nontemporal_load(&rps_n[o]));
        lse_merge(Mp, Sp, __builtin_nontemporal_load(&rpm_p[o]), __builtin_nontemporal_load(&rps_p[o]));
        pc += __builtin_nontemporal_load(&rpc[o]);     // carry-safe: cn sums < 2^16
    }
    rowLoss[row] = two_way_loss(Mn, Sn, Mp, Sp, pc >> 16, pc & 0xFFFF);
}

// ---------------------------------------------------------------------------
// Merge 16 row-chunk partials per column (fixed order) -> per-class loss.
// ---------------------------------------------------------------------------
__global__ __launch_bounds__(NBLK) void col_combine_kernel(
        const float* __restrict__ cpm_n, const float* __restrict__ cps_n,
        const float* __restrict__ cpm_p, const float* __restrict__ cps_p,
        const int* __restrict__ cpc, float* __restrict__ colLoss) {
    const int col = blockIdx.x * NBLK + threadIdx.x;
    if (col >= C_COLS) return;
    float Mn = FLT_BIG_NEG, Sn = 0.0f, Mp = FLT_BIG_NEG, Sp = 0.0f;
    int pc = 0;
#pragma unroll
    for (int ch = 0; ch < NCHUNK; ++ch) {
        const size_t o = (size_t)ch * C_COLS + col;
        lse_merge(Mn, Sn, __builtin_nontemporal_load(&cpm_n[o]), __builtin_nontemporal_load(&cps_n[o]));
        lse_merge(Mp, Sp, __builtin_nontemporal_load(&cpm_p[o]), __builtin_nontemporal_load(&cps_p[o]));
        pc += __builtin_nontemporal_load(&cpc[o]);
    }
    colLoss[col] = two_way_loss(Mn, Sn, Mp, Sp, pc >> 16, pc & 0xFFFF);
}

// ---------------------------------------------------------------------------
// Deterministic final reduce -> scalar.
// ---------------------------------------------------------------------------
__global__ __launch_bounds__(NBLK) void final_kernel(const float* __restrict__ rowLoss,
                                                     const float* __restrict__ colLoss,
                                                     float* __restrict__ out) {
    float sr = 0.0f, sc = 0.0f;
    for (int i = threadIdx.x; i < B_ROWS; i += NBLK) sr += rowLoss[i];
    for (int i = threadIdx.x; i < C_COLS; i += NBLK) sc += colLoss[i];
#pragma unroll
    for (int off = 16; off > 0; off >>= 1) {
        sr += __shfl_xor(sr, off);
        sc += __shfl_xor(sc, off);
    }
    __shared__ float ar[8], ac[8];
    const int wave = threadIdx.x >> 5, lane = threadIdx.x & 31;
    if (lane == 0) { ar[wave] = sr; ac[wave] = sc; }
    __syncthreads();
    if (threadIdx.x == 0) {
        float SR = 0.0f, SC = 0.0f;
#pragma unroll
        for (int w = 0; w < 8; ++w) { SR += ar[w]; SC += ac[w]; }
        out[0] = 0.5f * (SR / (float)B_ROWS + SC / (float)C_COLS);
    }
}

extern "C" void kernel_launch(void* const* d_in, const int* in_sizes, int n_in,
                              void* d_out, int out_size, void* d_ws, size_t ws_size,
                              hipStream_t stream) {
    (void)in_sizes; (void)n_in; (void)out_size; (void)ws_size;
    const float* logits  = (const float*)d_in[0];
    const int*   targets = (const int*)d_in[1];
    float* out = (float*)d_out;

    // Workspace layout (floats/ints, 4B each). Total ~818,816 slots ~= 3.3 MB.
    float* ws      = (float*)d_ws;
    float* rowLoss = ws;                       // 4096
    float* colLoss = ws + B_ROWS;              // 5000 (next free 9096; pad to 9216)
    const size_t RP = (size_t)CCHUNKS * B_ROWS;  // 81920 per array
    const size_t CP = (size_t)NCHUNK * C_COLS;   // 80000 per array
    float* rpm_n = ws + 9216;
    float* rps_n = rpm_n + RP;
    float* rpm_p = rps_n + RP;
    float* rps_p = rpm_p + RP;
    int*   rpc   = (int*)(rps_p + RP);
    float* cpm_n = (float*)(rpc + RP);
    float* cps_n = cpm_n + CP;
    float* cpm_p = cps_n + CP;
    float* cps_p = cpm_p + CP;
    int*   cpc   = (int*)(cps_p + CP);

    dim3 gf(CCHUNKS, NCHUNK);  // 20 x 16 = 320 blocks, one 256x256 tile each
    fused_partial_kernel<<<gf, NBLK, 0, stream>>>(logits, targets,
                                                  rpm_n, rps_n, rpm_p, rps_p, rpc,
                                                  cpm_n, cps_n, cpm_p, cps_p, cpc);

    row_combine_kernel<<<B_ROWS / NBLK, NBLK, 0, stream>>>(rpm_n, rps_n, rpm_p, rps_p, rpc, rowLoss);
    col_combine_kernel<<<(C_COLS + NBLK - 1) / NBLK, NBLK, 0, stream>>>(cpm_n, cps_n, cpm_p, cps_p, cpc, colLoss);
    final_kernel<<<1, NBLK, 0, stream>>>(rowLoss, colLoss, out);
}